// DynamicRangeCompression_31267361915320
// MI455X (gfx1250) — compile-verified
//
#include <hip/hip_runtime.h>

// ---------------------------------------------------------------------------
// DynamicRangeCompression for MI455X (gfx1250)
//   Kernel 1: elementwise target-gain f = 10^(gdb/20)  (v_log_f32 / v_exp_f32)
//             -> written into d_out (reused as f buffer)
//   Kernel 2: serial compressor scan, 1 workgroup, 128 lanes (1 lane = 1 row),
//             TDM (tensor_load_to_lds) triple-buffered strided tiles -> LDS,
//             with TDM padding for bank-conflict-free reads; overwrites d_out
//             in place with x * g * makeup.
// ---------------------------------------------------------------------------

#define BROWS   128          // batch rows (fixed by reference)
#define CHUNK   64           // timesteps per TDM tile
#define PITCH   68           // CHUNK + 4 dwords of TDM pad (LDS row pitch)
#define TILEF   (BROWS * PITCH)      // floats per tile (8704)
#define TILEB   (TILEF * 4)          // bytes per tile  (34816)
#define NBUF    3                    // triple buffering

typedef unsigned int v4u __attribute__((ext_vector_type(4)));
typedef int          v8i __attribute__((ext_vector_type(8)));
typedef int          v4i __attribute__((ext_vector_type(4)));
typedef float        f4v __attribute__((ext_vector_type(4)));

// Device pass: raw v_log_f32 / v_exp_f32. Host pass (semantic check only):
// standard libm names that have host declarations.
#if defined(__HIP_DEVICE_COMPILE__) && __has_builtin(__builtin_amdgcn_logf)
#define FAST_LOG2(v) __builtin_amdgcn_logf(v)
#else
#define FAST_LOG2(v) log2f(v)
#endif
#if defined(__HIP_DEVICE_COMPILE__) && __has_builtin(__builtin_amdgcn_exp2f)
#define FAST_EXP2(v) __builtin_amdgcn_exp2f(v)
#else
#define FAST_EXP2(v) exp2f(v)
#endif

// ---------------------------------------------------------------------------
// Kernel 1: target gain (fully parallel, memory bound: ~134 MB -> ~6 us)
//   y_abs = |x| + 1e-10 ;  log_x = 20*log10(y_abs)
//   gdb   = min(comp_slope*(thr - log_x), exp_slope*(-1000 - log_x), 0)
//   f     = 2^(gdb * log2(10)/20)
// ---------------------------------------------------------------------------
__device__ __forceinline__ float gain1(float xv, float cs, float cth, float es) {
    float v   = fabsf(xv) + 1e-10f;
    float lg  = 6.02059991327962f * FAST_LOG2(v);      // 20*log10(v)
    float a   = cs * (cth - lg);
    float b   = es * (-1000.0f - lg);
    float gdb = fminf(fminf(a, b), 0.0f);
    return FAST_EXP2(gdb * 0.16609640474436813f);      // 10^(gdb/20)
}

__global__ void __launch_bounds__(256)
gain_kernel(const float* __restrict__ x, const float* __restrict__ p,
            float* __restrict__ f, int n4) {
    int i = blockIdx.x * blockDim.x + threadIdx.x;
    if (i >= n4) return;
    const float cth = p[0];
    const float cs  = 1.0f - 1.0f / p[1];
    const float es  = 1.0f - 1.0f / 1e-10f;            // expander slope (~ -1e10)
    f4v xv = reinterpret_cast<const f4v*>(x)[i];
    f4v g;
    g.x = gain1(xv.x, cs, cth, es);
    g.y = gain1(xv.y, cs, cth, es);
    g.z = gain1(xv.z, cs, cth, es);
    g.w = gain1(xv.w, cs, cth, es);
    reinterpret_cast<f4v*>(f)[i] = g;
}

// ---------------------------------------------------------------------------
// TDM descriptor build + issue (ISA cdna5 §8.3/8.4; amdgpu-toolchain 6-arg
// builtin: groups 0..3 + extra zero-filled int32x8 + cache policy):
//   2D tile: tile_dim0 = CHUNK cols, tile_dim1 = 128 rows,
//   tensor_dim0_stride = T elements, data_size = 4B,
//   pad_enable: 4 dwords every 64 dwords -> LDS row pitch 68 dwords.
// ---------------------------------------------------------------------------
__device__ __forceinline__ void tdm_load_tile(const float* gtile,
                                              unsigned lds_off, unsigned T) {
    unsigned long long ga = (unsigned long long)gtile;
    v4u g0;
    g0.x = 1u;                                              // count=1, user mode
    g0.y = lds_off;                                         // lds_addr
    g0.z = (unsigned)(ga & 0xffffffffull);                  // global_addr lo
    g0.w = (unsigned)((ga >> 32) & 0x01ffffffull) | (2u << 30); // ga hi | type=2
    v8i g1;
    g1[0] = (int)((2u << 16)        // data_size = 4B
                | (1u << 20)        // pad_enable
                | (5u << 22)        // pad_interval: 64 dwords
                | (3u << 25));      // pad_amount: 4 dwords
    g1[1] = (int)((T & 0xffffu) << 16);                           // tdim0 lo16
    g1[2] = (int)(((T >> 16) & 0xffffu) | ((unsigned)BROWS << 16)); // tdim0 hi | tdim1 lo
    g1[3] = (int)(((unsigned)CHUNK) << 16);                       // tdim1 hi=0 | tile_dim0
    g1[4] = (int)((unsigned)BROWS);                               // tile_dim1 | tile_dim2=0
    g1[5] = (int)T;                                               // dim0_stride lo32
    g1[6] = 0;                                                    // stride hi | dim1_stride lo
    g1[7] = 0;
    v4i z4 = {0, 0, 0, 0};
    v8i z8 = {0, 0, 0, 0, 0, 0, 0, 0};
#if defined(__HIP_DEVICE_COMPILE__)
    __builtin_amdgcn_tensor_load_to_lds(g0, g1, z4, z4, z8, 0);
#else
    (void)g0; (void)g1; (void)z4; (void)z8;
#endif
}

__device__ __forceinline__ void wait_tensorcnt0() {
#if defined(__HIP_DEVICE_COMPILE__)
    __builtin_amdgcn_s_wait_tensorcnt(0);
#endif
}
__device__ __forceinline__ void wait_tensorcnt2() {
#if defined(__HIP_DEVICE_COMPILE__)
    __builtin_amdgcn_s_wait_tensorcnt(2);
#endif
}
__device__ __forceinline__ void wait_tensorcnt4() {
#if defined(__HIP_DEVICE_COMPILE__)
    __builtin_amdgcn_s_wait_tensorcnt(4);
#endif
}

// ---------------------------------------------------------------------------
// Kernel 2: sequential compressor core. One workgroup, 128 threads (4 waves).
// Lane b owns row b. fbuf (== d_out) holds target gains; overwritten in place.
// ---------------------------------------------------------------------------
#define STEP(F, X, O)                                            \
    do {                                                         \
        float f_  = (F);                                         \
        float d_  = f_ - y;                                      \
        float ya_ = fmaf(at, d_, y);                             \
        float yr_ = fmaf(rt, d_, y);                             \
        y = (f_ < y) ? ya_ : yr_;                                \
        (O) = (X) * y * mk;                                      \
    } while (0)

__global__ void __launch_bounds__(128)
compressor_scan_kernel(const float* __restrict__ x, float* __restrict__ fbuf,
                       const float* __restrict__ p, int T) {
    extern __shared__ float smem[];   // [NBUF f-tiles][NBUF x-tiles], 6*34816 B
    const int b = threadIdx.x;        // row index 0..127
    const unsigned ldsBase = (unsigned)(unsigned long long)smem;
    const float at = p[2];
    const float rt = p[3];
    const float mk = p[4];
    const int nchunks = T / CHUNK;
    const bool wave0 = (threadIdx.x < 32);   // wave-uniform predicate

    // Prologue: prefetch chunks 0..2 (f and x tiles).
    if (wave0) {
        for (int k = 0; k < NBUF && k < nchunks; ++k) {
            tdm_load_tile(fbuf + (size_t)k * CHUNK,
                          ldsBase + (unsigned)k * TILEB, (unsigned)T);
            tdm_load_tile(x + (size_t)k * CHUNK,
                          ldsBase + (unsigned)(NBUF + k) * TILEB, (unsigned)T);
        }
    }

    float y = 1.0f;   // zi = ones
    for (int k = 0; k < nchunks; ++k) {
        if (wave0) {
            if (k < nchunks - 2)      wait_tensorcnt4();
            else if (k < nchunks - 1) wait_tensorcnt2();
            else                      wait_tensorcnt0();
        }
        __syncthreads();                 // chunk k resident in LDS for all waves
        asm volatile("" ::: "memory");   // TDM wrote LDS behind compiler's back

        const int buf = k % NBUF;
        const f4v* fv =
            reinterpret_cast<const f4v*>(smem + buf * TILEF + b * PITCH);
        const f4v* xv =
            reinterpret_cast<const f4v*>(smem + (NBUF + buf) * TILEF + b * PITCH);
        f4v* orow =
            reinterpret_cast<f4v*>(fbuf + (size_t)b * T + (size_t)k * CHUNK);

#pragma unroll
        for (int t4 = 0; t4 < CHUNK / 4; ++t4) {
            f4v f4 = fv[t4];             // ds_load_b128 (pitch 68 dwords)
            f4v x4 = xv[t4];
            f4v o4;
            STEP(f4.x, x4.x, o4.x);
            STEP(f4.y, x4.y, o4.y);
            STEP(f4.z, x4.z, o4.z);
            STEP(f4.w, x4.w, o4.w);
            orow[t4] = o4;               // global_store_b128, per-lane row
        }
        __syncthreads();                 // everyone done reading buffer k%NBUF

        if (wave0 && (k + NBUF) < nchunks) {  // refill just-freed buffer
            tdm_load_tile(fbuf + (size_t)(k + NBUF) * CHUNK,
                          ldsBase + (unsigned)buf * TILEB, (unsigned)T);
            tdm_load_tile(x + (size_t)(k + NBUF) * CHUNK,
                          ldsBase + (unsigned)(NBUF + buf) * TILEB, (unsigned)T);
        }
    }
}

// ---------------------------------------------------------------------------
// Host launcher
// ---------------------------------------------------------------------------
extern "C" void kernel_launch(void* const* d_in, const int* in_sizes, int n_in,
                              void* d_out, int out_size, void* d_ws, size_t ws_size,
                              hipStream_t stream) {
    const float* x      = (const float*)d_in[0];
    const float* params = (const float*)d_in[1];
    float* out          = (float*)d_out;

    const int N = in_sizes[0];        // B*T = 128 * 131072
    const int T = N / BROWS;

    // Pass 1: target gains into d_out (memory bound, ~6 us)
    const int n4 = N / 4;
    gain_kernel<<<(n4 + 255) / 256, 256, 0, stream>>>(x, params, out, n4);

    // Pass 2: serial scan, single workgroup, TDM triple-buffered LDS tiles
    const size_t shbytes = (size_t)(2 * NBUF) * TILEB;   // 208,896 bytes
    compressor_scan_kernel<<<1, BROWS, shbytes, stream>>>(x, out, params, T);
}